// LeakDetector_21973052686651
// MI455X (gfx1250) — compile-verified
//
#include <hip/hip_runtime.h>
#include <math.h>

// ---------------------------------------------------------------------------
// GAT autoencoder for MI455X (gfx1250, wave32).
// Dense GEMMs use V_WMMA_F32_16X16X4_F32 (full fp32, matches reference).
// Edge phase: float-keyed atomicMax segment-softmax + wave-per-edge scatter.
// ---------------------------------------------------------------------------

typedef float v2f __attribute__((ext_vector_type(2)));
typedef float v8f __attribute__((ext_vector_type(8)));

#define NEG_SLOPE 0.2f
#define BN_EPS 1e-5f

// Monotone float -> uint key (order-preserving) for atomicMax-based segment max
__device__ __forceinline__ unsigned fkey(float f) {
  unsigned u = __float_as_uint(f);
  return (u & 0x80000000u) ? ~u : (u | 0x80000000u);
}
__device__ __forceinline__ float funkey(unsigned u) {
  u = (u & 0x80000000u) ? (u ^ 0x80000000u) : ~u;
  return __uint_as_float(u);
}

// ---------------------------------------------------------------------------
// GEMM: C[n,F] = A[n,K] @ B[K,F], one wave per 16x16 tile, f32 WMMA.
// A-frag 16x4: lane<16 -> K+0/K+1 in a[0]/a[1]; lane>=16 -> K+2/K+3.
// B-frag 4x16: b[0] = B[K+(half?2:0), col], b[1] = B[K+(half?3:1), col].
// C/D: acc[i] is row (mt*16 + i + 8*half), col (nt*16 + lane&15).
// Row index is CLAMPED (not masked) so loads never diverge and EXEC stays
// full for WMMA; out-of-range output rows are dropped at the store (WMMA
// output row M depends only on A row M, so clamping cannot pollute valid rows).
// ---------------------------------------------------------------------------
__global__ void k_gemm_wmma(const float* __restrict__ A, const float* __restrict__ B,
                            float* __restrict__ C, int n, int K, int F,
                            int mtiles, int ntiles) {
  int tile = blockIdx.x * (blockDim.x >> 5) + (threadIdx.x >> 5);
  if (tile >= mtiles * ntiles) return;  // wave-uniform: EXEC stays full for WMMA
  int mt = tile / ntiles, nt = tile % ntiles;
  int lane = threadIdx.x & 31;
  int half = lane >> 4;
  int l15  = lane & 15;
  int row = mt * 16 + l15;
  int rowc = row < n ? row : (n - 1);   // clamp: keeps loads in-bounds, no divergence
  int col = nt * 16 + l15;
  v8f acc = {0.f, 0.f, 0.f, 0.f, 0.f, 0.f, 0.f, 0.f};

  const float* ap = A + (size_t)rowc * K + 2 * half;       // 8B-aligned
  const float* bp = B + (size_t)(2 * half) * F + col;
  long bstride = (long)F;

#if __has_builtin(__builtin_amdgcn_wmma_f32_16x16x4_f32)
  for (int k = 0; k < K; k += 4) {
    v2f a = *(const v2f*)ap;            // global_load_b64
    v2f b;
    b[0] = bp[0];
    b[1] = bp[bstride];
    acc = __builtin_amdgcn_wmma_f32_16x16x4_f32(
        /*neg_a=*/false, a, /*neg_b=*/false, b,
        /*c_mod=*/(short)0, acc, /*reuse_a=*/false, /*reuse_b=*/false);
    ap += 4;
    bp += 4 * bstride;
  }
#else
  {
    int bm = mt * 16 + half * 8;
    for (int i = 0; i < 8; ++i) {
      int r = bm + i;
      float s = 0.f;
      if (r < n)
        for (int k = 0; k < K; ++k) s += A[(size_t)r * K + k] * B[(size_t)k * F + col];
      acc[i] = s;
    }
  }
#endif

  int bm = mt * 16 + half * 8;
  for (int i = 0; i < 8; ++i)
    if (bm + i < n) C[(size_t)(bm + i) * F + col] = acc[i];
}

// ---------------------------------------------------------------------------
__global__ void k_fill(float* __restrict__ p, size_t count, float v) {
  size_t i = (size_t)blockIdx.x * blockDim.x + threadIdx.x;
  if (i < count) p[i] = v;
}

// Per (node, head): e_src/e_dst dot products; init segment-max key and denom.
// Hh is a power of two (1 or 4): hshift = log2(Hh).
__global__ void k_node_prep(const float* __restrict__ HB,
                            const float* __restrict__ a_src,
                            const float* __restrict__ a_dst,
                            float* __restrict__ es, float* __restrict__ ed,
                            unsigned* __restrict__ mkey, float* __restrict__ den,
                            int n, int Hh, int D, int hshift) {
  int i = blockIdx.x * blockDim.x + threadIdx.x;
  if (i >= (n << hshift)) return;
  int node = i >> hshift;
  int h = i & (Hh - 1);
  const float* hp = HB + ((size_t)node << hshift) * D + (size_t)h * D;
  float s1 = 0.f, s2 = 0.f;
  for (int d = 0; d < D; ++d) {
    float v = hp[d];
    s1 += v * a_src[h * D + d];
    s2 += v * a_dst[h * D + d];
  }
  es[i] = s1;
  ed[i] = s2;
  mkey[i] = 0u;   // below any real float key
  den[i] = 0.f;
}

// Pass 1 over edges: alpha = leaky_relu(e_src[s]+e_dst[d]); store; atomicMax per dst.
__global__ void k_edge_alpha(const int* __restrict__ src, const int* __restrict__ dst,
                             const float* __restrict__ es, const float* __restrict__ ed,
                             float* __restrict__ alpha, unsigned* __restrict__ mkey,
                             int E, int n, int Hh, int hshift) {
  int e = blockIdx.x * blockDim.x + threadIdx.x;
  int tot = E + n;
  if (e >= tot) return;
  int s = (e < E) ? src[e] : (e - E);
  int d = (e < E) ? dst[e] : (e - E);
  int sb = s << hshift, db = d << hshift;
  size_t eb = (size_t)e << hshift;
  for (int h = 0; h < Hh; ++h) {
    float a = es[sb + h] + ed[db + h];
    a = (a > 0.f) ? a : NEG_SLOPE * a;
    alpha[eb + h] = a;
    atomicMax(&mkey[db + h], fkey(a));
  }
}

// Pass 2: ex = exp(alpha - max[dst]); overwrite alpha with ex; atomicAdd denom.
__global__ void k_edge_exp(const int* __restrict__ dst,
                           float* __restrict__ alpha, const unsigned* __restrict__ mkey,
                           float* __restrict__ den, int E, int n, int Hh, int hshift) {
  int e = blockIdx.x * blockDim.x + threadIdx.x;
  int tot = E + n;
  if (e >= tot) return;
  int d = (e < E) ? dst[e] : (e - E);
  int db = d << hshift;
  size_t eb = (size_t)e << hshift;
  for (int h = 0; h < Hh; ++h) {
    float a = alpha[eb + h];
    float ex = __expf(a - funkey(mkey[db + h]));
    alpha[eb + h] = ex;
    atomicAdd(&den[db + h], ex);
  }
}

// Pass 3: one wave per edge; lanes stride features; out[dst] += (ex/den)*h[src].
// D is a power of two (32 or 128): dshift = log2(D).
__global__ void k_edge_scatter(const int* __restrict__ src, const int* __restrict__ dst,
                               const float* __restrict__ ex, const float* __restrict__ den,
                               const float* __restrict__ HB, float* __restrict__ OUT,
                               int E, int n, int Hh, int D, int dshift, int hshift) {
  int wid = (int)(((size_t)blockIdx.x * blockDim.x + threadIdx.x) >> 5);
  int lane = threadIdx.x & 31;
  int tot = E + n;
  if (wid >= tot) return;
  int s = (wid < E) ? src[wid] : (wid - E);
  int d = (wid < E) ? dst[wid] : (wid - E);
  int F = Hh << dshift;
  size_t eb = (size_t)wid << hshift;
  size_t sb = (size_t)s * F;
  size_t db = (size_t)d * F;
  int dbh = d << hshift;
  for (int f = lane; f < F; f += 32) {
    int h = f >> dshift;
    float w = ex[eb + h] / den[dbh + h];
    atomicAdd(&OUT[db + f], w * HB[sb + f]);
  }
}

// Epilogue: x += bias (per feature); optional relu. F power of two.
__global__ void k_bias_act(float* __restrict__ X, const float* __restrict__ b,
                           int n, int F, int do_relu) {
  size_t i = (size_t)blockIdx.x * blockDim.x + threadIdx.x;
  if (i >= (size_t)n * F) return;
  float v = X[i] + b[i & (F - 1)];
  if (do_relu) v = fmaxf(v, 0.f);
  X[i] = v;
}

// BatchNorm stats: one block per feature; compute scale/shift.
__global__ void k_bn_stats(const float* __restrict__ X, const float* __restrict__ g,
                           const float* __restrict__ be, float* __restrict__ scale,
                           float* __restrict__ shift, int n, int F) {
  int f = blockIdx.x;
  int t = threadIdx.x;
  __shared__ float ss[256];
  __shared__ float sq[256];
  float s = 0.f, q = 0.f;
  for (int i = t; i < n; i += 256) {
    float v = X[(size_t)i * F + f];
    s += v;
    q += v * v;
  }
  ss[t] = s; sq[t] = q;
  __syncthreads();
  for (int o = 128; o > 0; o >>= 1) {
    if (t < o) { ss[t] += ss[t + o]; sq[t] += sq[t + o]; }
    __syncthreads();
  }
  if (t == 0) {
    float inv_n = 1.f / (float)n;
    float mu = ss[0] * inv_n;
    float var = sq[0] * inv_n - mu * mu;
    float sc = g[f] * rsqrtf(var + BN_EPS);
    scale[f] = sc;
    shift[f] = be[f] - mu * sc;
  }
}

__global__ void k_bn_apply(float* __restrict__ X, const float* __restrict__ scale,
                           const float* __restrict__ shift, int n, int F, int relu_after) {
  size_t i = (size_t)blockIdx.x * blockDim.x + threadIdx.x;
  if (i >= (size_t)n * F) return;
  int f = (int)(i & (F - 1));
  float v = X[i] * scale[f] + shift[f];
  if (relu_after) v = fmaxf(v, 0.f);
  X[i] = v;
}

// ---------------------------------------------------------------------------
// Host-side layer drivers
// ---------------------------------------------------------------------------
static inline int ilog2(int v) { int s = 0; while ((1 << s) < v) ++s; return s; }

static void gat_layer(const float* X, int Fin, const float* W, const float* as_,
                      const float* ad_, const float* bias, int Hh, int D,
                      float* OUT, int relu_flag,
                      float* HB, float* ES, float* ED, unsigned* MB, float* DEN,
                      float* ALPHA, const int* src, const int* dst,
                      int n, int E, hipStream_t stream) {
  int F = Hh * D;
  int hshift = ilog2(Hh), dshift = ilog2(D);
  int mtiles = (n + 15) / 16, ntiles = F / 16;
  int tiles = mtiles * ntiles;
  k_gemm_wmma<<<(tiles + 7) / 8, 256, 0, stream>>>(X, W, HB, n, Fin, F, mtiles, ntiles);
  size_t nf = (size_t)n * F;
  k_fill<<<(unsigned)((nf + 255) / 256), 256, 0, stream>>>(OUT, nf, 0.f);
  k_node_prep<<<(n * Hh + 255) / 256, 256, 0, stream>>>(HB, as_, ad_, ES, ED, MB, DEN,
                                                        n, Hh, D, hshift);
  int tot = E + n;
  k_edge_alpha<<<(tot + 255) / 256, 256, 0, stream>>>(src, dst, ES, ED, ALPHA, MB,
                                                      E, n, Hh, hshift);
  k_edge_exp<<<(tot + 255) / 256, 256, 0, stream>>>(dst, ALPHA, MB, DEN, E, n, Hh, hshift);
  size_t sthreads = (size_t)tot * 32;
  k_edge_scatter<<<(unsigned)((sthreads + 255) / 256), 256, 0, stream>>>(
      src, dst, ALPHA, DEN, HB, OUT, E, n, Hh, D, dshift, hshift);
  k_bias_act<<<(unsigned)((nf + 255) / 256), 256, 0, stream>>>(OUT, bias, n, F, relu_flag);
}

static void bn_layer(float* X, const float* g, const float* be, float* SC, float* SH,
                     int n, int F, int relu_after, hipStream_t stream) {
  k_bn_stats<<<F, 256, 0, stream>>>(X, g, be, SC, SH, n, F);
  size_t nf = (size_t)n * F;
  k_bn_apply<<<(unsigned)((nf + 255) / 256), 256, 0, stream>>>(X, SC, SH, n, F, relu_after);
}

extern "C" void kernel_launch(void* const* d_in, const int* in_sizes, int n_in,
                              void* d_out, int out_size, void* d_ws, size_t ws_size,
                              hipStream_t stream) {
  const float* x  = (const float*)d_in[0];
  const int* ei   = (const int*)d_in[1];
  const float* W1 = (const float*)d_in[2];  const float* as1 = (const float*)d_in[3];
  const float* ad1= (const float*)d_in[4];  const float* b1  = (const float*)d_in[5];
  const float* g1 = (const float*)d_in[6];  const float* be1 = (const float*)d_in[7];
  const float* W2 = (const float*)d_in[8];  const float* as2 = (const float*)d_in[9];
  const float* ad2= (const float*)d_in[10]; const float* b2  = (const float*)d_in[11];
  const float* g2 = (const float*)d_in[12]; const float* be2 = (const float*)d_in[13];
  const float* W3 = (const float*)d_in[14]; const float* as3 = (const float*)d_in[15];
  const float* ad3= (const float*)d_in[16]; const float* b3  = (const float*)d_in[17];
  const float* W4 = (const float*)d_in[18]; const float* as4 = (const float*)d_in[19];
  const float* ad4= (const float*)d_in[20]; const float* b4  = (const float*)d_in[21];
  const float* g4 = (const float*)d_in[22]; const float* be4 = (const float*)d_in[23];
  const float* W5 = (const float*)d_in[24]; const float* as5 = (const float*)d_in[25];
  const float* ad5= (const float*)d_in[26]; const float* b5  = (const float*)d_in[27];

  const int NODE_IN = 128, HID = 32;
  int n = in_sizes[0] / NODE_IN;   // 50000
  int E = in_sizes[1] / 2;         // 800000
  const int* src = ei;
  const int* dst = ei + E;

  // Workspace carve-up (floats)
  float* ws = (float*)d_ws;
  size_t off = 0;
  float* P0 = ws + off;     off += (size_t)n * NODE_IN;
  float* P1 = ws + off;     off += (size_t)n * NODE_IN;
  float* HB = ws + off;     off += (size_t)n * NODE_IN;
  float* ES = ws + off;     off += (size_t)n * 4;
  float* ED = ws + off;     off += (size_t)n * 4;
  unsigned* MB = (unsigned*)(ws + off); off += (size_t)n * 4;
  float* DEN = ws + off;    off += (size_t)n * 4;
  float* ALPHA = ws + off;  off += (size_t)(E + n) * 4;
  float* SC = ws + off;     off += NODE_IN;
  float* SH = ws + off;     off += NODE_IN;
  (void)ws_size; (void)n_in; (void)out_size;

  // L1: 128 -> 4x32, relu, bn
  gat_layer(x, NODE_IN, W1, as1, ad1, b1, 4, HID, P0, 1,
            HB, ES, ED, MB, DEN, ALPHA, src, dst, n, E, stream);
  bn_layer(P0, g1, be1, SC, SH, n, 4 * HID, 0, stream);

  // L2: 128 -> 4x32, relu, bn
  gat_layer(P0, 4 * HID, W2, as2, ad2, b2, 4, HID, P1, 1,
            HB, ES, ED, MB, DEN, ALPHA, src, dst, n, E, stream);
  bn_layer(P1, g2, be2, SC, SH, n, 4 * HID, 0, stream);

  // L3 (bottleneck): 128 -> 32, relu
  gat_layer(P1, 4 * HID, W3, as3, ad3, b3, 1, HID, P0, 1,
            HB, ES, ED, MB, DEN, ALPHA, src, dst, n, E, stream);

  // L4: 32 -> 32, relu, bn, relu
  gat_layer(P0, HID, W4, as4, ad4, b4, 1, HID, P1, 1,
            HB, ES, ED, MB, DEN, ALPHA, src, dst, n, E, stream);
  bn_layer(P1, g4, be4, SC, SH, n, HID, 1, stream);

  // L5: 32 -> 128, no relu, write directly to d_out
  gat_layer(P1, HID, W5, as5, ad5, b5, 1, NODE_IN, (float*)d_out, 0,
            HB, ES, ED, MB, DEN, ALPHA, src, dst, n, E, stream);
}